// SmartPostProcess_31765578121485
// MI455X (gfx1250) — compile-verified
//
#include <hip/hip_runtime.h>
#include <math.h>

// ---------------- problem constants ----------------
#define NN    192
#define NN2   (192 * 192)
#define VOX   (192 * 192 * 192)          // 7,077,888
#define NVOL  2
#define TOTAL (NVOL * VOX)               // 14,155,776
#define VOXBLOCKS (TOTAL / 256)          // 55,296
#define NNW   (NN / 4)                   // 48 words per x-row
#define NN2W  (NN2 / 4)
#define VOXW  (VOX / 4)
#define TOTALW (TOTAL / 4)               // 3,538,944
#define WORDBLOCKS (TOTALW / 256)        // 13,824
#define CONVWAVES (NVOL * NN * 144)      // 55,296 waves (one 16x16 tile each)
#define CONVBLOCKS (CONVWAVES / 4)       // 13,824 blocks of 128 threads

typedef __attribute__((ext_vector_type(2))) float v2f;
typedef __attribute__((ext_vector_type(8))) float v8f;

// ---------------- helpers ----------------
__device__ __forceinline__ void decode_voxel(int t, int& vol, int& rem,
                                             int& z, int& y, int& x) {
  vol = t / VOX;
  rem = t - vol * VOX;
  z = rem / NN2;
  int r2 = rem - z * NN2;
  y = r2 / NN;
  x = r2 - y * NN;
}

// Branch-free guarded loads: clamp address (always valid), cndmask the value.
__device__ __forceinline__ float load_or_zero_f(const float* b, int idx, int off, bool ok) {
  float v = b[ok ? idx + off : idx];
  return ok ? v : 0.0f;
}
__device__ __forceinline__ int load_or_zero_i(const int* b, int idx, int off, bool ok) {
  int v = b[ok ? idx + off : idx];
  return ok ? v : 0;
}
__device__ __forceinline__ unsigned load_or_zero_u(const unsigned* b, int idx, int off, bool ok) {
  unsigned v = b[ok ? idx + off : idx];
  return ok ? v : 0u;
}

// Band tap lookup from LDS (clamped index + cndmask, no branches, no exp).
template <int R>
__device__ __forceinline__ float band_tap(const float* staps, int t) {
  const unsigned ut = (unsigned)t;
  const bool ok = ut <= (unsigned)(2 * R);
  float v = staps[ok ? ut : 0u];
  return ok ? v : 0.0f;
}

// Thread 0 of the block fills normalized Gaussian taps into LDS.
template <int R>
__device__ __forceinline__ void fill_taps(float* staps, float sigma) {
  if (threadIdx.x == 0) {
    const float inv2s2 = 1.0f / (2.0f * sigma * sigma);
    float tmp[2 * R + 1];
    float s = 0.0f;
#pragma unroll
    for (int t = 0; t <= 2 * R; ++t) {
      float d = (float)(t - R);
      float e = __expf(-d * d * inv2s2);
      tmp[t] = e;
      s += e;
    }
    const float inv = 1.0f / s;
#pragma unroll
    for (int t = 0; t <= 2 * R; ++t) staps[t] = tmp[t] * inv;
  }
  __syncthreads();
}

// ============================================================
// Separable Gaussian conv along X (contiguous axis) via WMMA.
// One wave computes a 16(y) x 16(x) tile at fixed z.
// out = A(16 x C) * Band(C x 16),  A = input patch, C = 16+2R padded to 4.
// ============================================================
template <int R>
__global__ __launch_bounds__(128) void conv_x_wmma(
    const float* __restrict__ src, float* __restrict__ dst, float sigma) {
  constexpr int C = 16 + 2 * R;
  constexpr int KC = (C + 3) / 4;

  __shared__ float staps[16];
  fill_taps<R>(staps, sigma);

  const int lane = threadIdx.x & 31;
  const int wave = threadIdx.x >> 5;
  const int w = blockIdx.x * 4 + wave;

  const int vol = w / (NN * 144);
  int rem = w - vol * (NN * 144);
  const int z = rem / 144;
  int tile = rem - z * 144;
  const int y0 = (tile / 12) * 16;
  const int x0 = (tile % 12) * 16;

  const float* sb = src + (size_t)vol * VOX + (size_t)z * NN2;
  float* db = dst + (size_t)vol * VOX + (size_t)z * NN2;

  const int half = lane >> 4;   // 0: lanes 0-15, 1: lanes 16-31
  const int li = lane & 15;
  const int rowbase = (y0 + li) * NN;

  v8f acc = {0.f, 0.f, 0.f, 0.f, 0.f, 0.f, 0.f, 0.f};
#pragma unroll
  for (int kc = 0; kc < KC; ++kc) {
    const int k0 = kc * 4 + 2 * half;
    const int k1 = k0 + 1;
    // A: input rows (y0+li), x columns x0-R+k (zero-padded, branch-free)
    const int xc0 = x0 - R + k0;
    const int xc1 = x0 - R + k1;
    const bool ok0 = (xc0 >= 0) && (xc0 < NN);
    const bool ok1 = (xc1 >= 0) && (xc1 < NN);
    v2f a;
    a.x = load_or_zero_f(sb, rowbase, xc0, ok0) ;
    a.y = load_or_zero_f(sb, rowbase, xc1, ok1);
    // B: band matrix B[k][c] = taps[k-c]
    v2f b;
    b.x = band_tap<R>(staps, k0 - li);
    b.y = band_tap<R>(staps, k1 - li);
    acc = __builtin_amdgcn_wmma_f32_16x16x4_f32(false, a, false, b,
                                                (short)0, acc, false, false);
  }
#pragma unroll
  for (int v = 0; v < 8; ++v) {
    const int M = v + 8 * half;          // D row (y)
    db[(y0 + M) * NN + x0 + li] = acc[v];
  }
}

// ============================================================
// Separable Gaussian conv along a strided axis (y or z) via WMMA.
// rows (M) = conv axis, cols (N) = x (contiguous -> coalesced B loads).
//   conv_y: rowStride = NN  (y), fixedStride = NN2 (z fixed)
//   conv_z: rowStride = NN2 (z), fixedStride = NN  (y fixed)
// ============================================================
template <int R>
__global__ __launch_bounds__(128) void conv_r_wmma(
    const float* __restrict__ src, float* __restrict__ dst, float sigma,
    int rowStride, int fixedStride) {
  constexpr int C = 16 + 2 * R;
  constexpr int KC = (C + 3) / 4;

  __shared__ float staps[16];
  fill_taps<R>(staps, sigma);

  const int lane = threadIdx.x & 31;
  const int wave = threadIdx.x >> 5;
  const int w = blockIdx.x * 4 + wave;

  const int vol = w / (NN * 144);
  int rem = w - vol * (NN * 144);
  const int fixedc = rem / 144;
  int tile = rem - fixedc * 144;
  const int r0 = (tile / 12) * 16;
  const int x0 = (tile % 12) * 16;

  const float* sb = src + (size_t)vol * VOX + (size_t)fixedc * fixedStride;
  float* db = dst + (size_t)vol * VOX + (size_t)fixedc * fixedStride;

  const int half = lane >> 4;
  const int li = lane & 15;

  v8f acc = {0.f, 0.f, 0.f, 0.f, 0.f, 0.f, 0.f, 0.f};
#pragma unroll
  for (int kc = 0; kc < KC; ++kc) {
    const int k0 = kc * 4 + 2 * half;
    const int k1 = k0 + 1;
    // A: band A[i][k] = taps[k-i], i = li
    v2f a;
    a.x = band_tap<R>(staps, k0 - li);
    a.y = band_tap<R>(staps, k1 - li);
    // B: input rows r0-R+k, contiguous x cols (coalesced, branch-free guards)
    const int rc0 = r0 - R + k0;
    const int rc1 = r0 - R + k1;
    const bool ok0 = (rc0 >= 0) && (rc0 < NN);
    const bool ok1 = (rc1 >= 0) && (rc1 < NN);
    v2f b;
    b.x = load_or_zero_f(sb, x0 + li, rc0 * rowStride, ok0);
    b.y = load_or_zero_f(sb, x0 + li, rc1 * rowStride, ok1);
    acc = __builtin_amdgcn_wmma_f32_16x16x4_f32(false, a, false, b,
                                                (short)0, acc, false, false);
  }
#pragma unroll
  for (int v = 0; v < 8; ++v) {
    const int M = v + 8 * half;
    db[(r0 + M) * rowStride + x0 + li] = acc[v];
  }
}

// ---------------- surfaceness from smoothed volume ----------------
__global__ __launch_bounds__(256) void surf_kernel(
    const float* __restrict__ s, float* __restrict__ surf, float scale) {
  const int t = blockIdx.x * 256 + threadIdx.x;
  int vol, rem, z, y, x;
  decode_voxel(t, vol, rem, z, y, x);
  const float* b = s + (size_t)vol * VOX;
  const float c = b[rem];
  const float zm = load_or_zero_f(b, rem, -NN2, z > 0);
  const float zp = load_or_zero_f(b, rem, NN2, z < NN - 1);
  const float ym = load_or_zero_f(b, rem, -NN, y > 0);
  const float yp = load_or_zero_f(b, rem, NN, y < NN - 1);
  const float xm = load_or_zero_f(b, rem, -1, x > 0);
  const float xp = load_or_zero_f(b, rem, 1, x < NN - 1);
  const float fxx = (zm - 2.f * c + zp) * scale;  // axis 0
  const float fyy = (ym - 2.f * c + yp) * scale;  // axis 1
  const float fzz = (xm - 2.f * c + xp) * scale;  // axis 2
  const float e = __expf(-(fxx * fxx + fyy * fyy) /
                         (0.25f * (fzz * fzz + 1e-7f)));
  surf[t] = e * fabsf(fzz);
}

// ---------------- per-volume max reduction ----------------
__global__ __launch_bounds__(256) void reduce_max_kernel(
    const float* __restrict__ surf, float* __restrict__ red) {
  __shared__ float sm[256];
  const int t = blockIdx.x * 256 + threadIdx.x;
  sm[threadIdx.x] = surf[t];          // surf >= 0, so uint-compare max is valid
  __syncthreads();
#pragma unroll
  for (int s = 128; s > 0; s >>= 1) {
    if (threadIdx.x < s) sm[threadIdx.x] = fmaxf(sm[threadIdx.x], sm[threadIdx.x + s]);
    __syncthreads();
  }
  if (threadIdx.x == 0) {
    const int vol = t / VOX;          // VOX % 256 == 0 -> block within one volume
    atomicMax((unsigned int*)(red + vol), __float_as_uint(sm[0]));
  }
}

__global__ void zero_u32_kernel(unsigned int* p, int n) {
  const int t = blockIdx.x * blockDim.x + threadIdx.x;
  if (t < n) p[t] = 0u;
}

// ---------------- normalize + max over sigmas ----------------
__global__ __launch_bounds__(256) void combine_kernel(
    const float* __restrict__ surf, const float* __restrict__ red,
    float* __restrict__ smax, int init) {
  const int t = blockIdx.x * 256 + threadIdx.x;
  const int vol = t / VOX;
  const float mx = red[vol];
  const float v = (mx > 0.f) ? surf[t] / (mx + 1e-7f) : 0.f;
  smax[t] = init ? v : fmaxf(smax[t], v);
}

// ---------------- thresholds ----------------
__global__ __launch_bounds__(256) void threshold_kernel(
    const float* __restrict__ vin, const float* __restrict__ smax,
    unsigned char* __restrict__ low, unsigned char* __restrict__ high) {
  const int t = blockIdx.x * 256 + threadIdx.x;
  const float val = vin[t] * smax[t];
  low[t] = (val > 0.5f) ? 1 : 0;
  high[t] = (val > 0.9f) ? 1 : 0;
}

// ---------------- hysteresis Jacobi step: 4 voxels/thread (u32 words) ----
// Bytes are 0/1 and only OR/AND are applied, so byte lanes cannot interact.
__global__ __launch_bounds__(256) void hyst4_kernel(
    const unsigned* __restrict__ low, const unsigned* __restrict__ src,
    unsigned* __restrict__ dst) {
  const int t4 = blockIdx.x * 256 + threadIdx.x;
  const int vol = t4 / VOXW;
  int rem = t4 - vol * VOXW;
  const int z = rem / NN2W;
  int r2 = rem - z * NN2W;
  const int y = r2 / NNW;
  const int xw = r2 - y * NNW;

  const unsigned c = src[t4];
  unsigned m = c | (c << 8) | (c >> 8);                       // in-word x+-1
  m |= load_or_zero_u(src, t4, -1, xw > 0) >> 24;             // left word byte3
  m |= load_or_zero_u(src, t4, 1, xw < NNW - 1) << 24;        // right word byte0
  m |= load_or_zero_u(src, t4, -NNW, y > 0);
  m |= load_or_zero_u(src, t4, NNW, y < NN - 1);
  m |= load_or_zero_u(src, t4, -NN2W, z > 0);
  m |= load_or_zero_u(src, t4, NN2W, z < NN - 1);
  dst[t4] = m & low[t4];
}

// ---------------- z-axis dilate/erode (radius 1), word-based ----------------
__global__ __launch_bounds__(256) void morph_z4_kernel(
    const unsigned* __restrict__ src, unsigned* __restrict__ dst, int dilate) {
  const int t4 = blockIdx.x * 256 + threadIdx.x;
  const int rem = t4 % VOXW;
  const int z = rem / NN2W;
  const unsigned pad = dilate ? 0u : 0x01010101u;
  const unsigned c = src[t4];
  unsigned a = src[(z > 0) ? t4 - NN2W : t4];
  a = (z > 0) ? a : pad;
  unsigned b = src[(z < NN - 1) ? t4 + NN2W : t4];
  b = (z < NN - 1) ? b : pad;
  dst[t4] = dilate ? (c | a | b) : (c & a & b);
}

// ---------------- connected components ----------------
__global__ __launch_bounds__(256) void cc_init_kernel(
    const unsigned char* __restrict__ mask, int* __restrict__ lab) {
  const int t = blockIdx.x * 256 + threadIdx.x;
  const int vol = t / VOX;
  const int rem = t - vol * VOX;
  lab[t] = mask[t] ? (rem + 1) : 0;
}

__global__ __launch_bounds__(256) void cc_step_kernel(
    const unsigned char* __restrict__ mask, const int* __restrict__ src,
    int* __restrict__ dst) {
  const int t = blockIdx.x * 256 + threadIdx.x;
  int vol, rem, z, y, x;
  decode_voxel(t, vol, rem, z, y, x);
  int l = src[t];
  l = max(l, load_or_zero_i(src, t, -NN2, z > 0));
  l = max(l, load_or_zero_i(src, t, NN2, z < NN - 1));
  l = max(l, load_or_zero_i(src, t, -NN, y > 0));
  l = max(l, load_or_zero_i(src, t, NN, y < NN - 1));
  l = max(l, load_or_zero_i(src, t, -1, x > 0));
  l = max(l, load_or_zero_i(src, t, 1, x < NN - 1));
  dst[t] = mask[t] ? l : 0;
}

__global__ __launch_bounds__(256) void cc_count_kernel(
    const unsigned char* __restrict__ mask, const int* __restrict__ lab,
    unsigned int* __restrict__ counts) {
  const int t = blockIdx.x * 256 + threadIdx.x;
  if (mask[t]) {
    const int vol = t / VOX;
    atomicAdd(&counts[(size_t)vol * (VOX + 1) + lab[t]], 1u);
  }
}

__global__ __launch_bounds__(256) void final_kernel(
    const unsigned char* __restrict__ mask, const int* __restrict__ lab,
    const unsigned int* __restrict__ counts, float* __restrict__ out) {
  const int t = blockIdx.x * 256 + threadIdx.x;
  float r = 0.0f;
  if (mask[t]) {
    const int vol = t / VOX;
    r = (counts[(size_t)vol * (VOX + 1) + lab[t]] >= 100u) ? 1.0f : 0.0f;
  }
  out[t] = r;
}

// ---------------- workspace layout (bytes) ----------------
static const size_t OFF_TMPA = 0;                                 // 2*VOX f32
static const size_t OFF_TMPB = OFF_TMPA + (size_t)TOTAL * 4;      // 2*VOX f32
static const size_t OFF_SMAX = OFF_TMPB + (size_t)TOTAL * 4;      // 2*VOX f32
static const size_t OFF_LOW  = OFF_SMAX + (size_t)TOTAL * 4;      // 2*VOX u8
static const size_t OFF_MA   = OFF_LOW + (size_t)TOTAL;           // 2*VOX u8
static const size_t OFF_MB   = OFF_MA + (size_t)TOTAL;            // 2*VOX u8
static const size_t OFF_RED  = OFF_MB + (size_t)TOTAL;            // 256 B
// counts aliases surfmax+low (both dead by CC time): 2*(VOX+1) u32 fits.
static const size_t OFF_CNT  = OFF_SMAX;

extern "C" void kernel_launch(void* const* d_in, const int* in_sizes, int n_in,
                              void* d_out, int out_size, void* d_ws, size_t ws_size,
                              hipStream_t stream) {
  (void)in_sizes; (void)n_in; (void)out_size; (void)ws_size;
  const float* vin = (const float*)d_in[0];
  float* out = (float*)d_out;
  char* ws = (char*)d_ws;

  float* tmpA = (float*)(ws + OFF_TMPA);
  float* tmpB = (float*)(ws + OFF_TMPB);
  float* smax = (float*)(ws + OFF_SMAX);
  unsigned char* low = (unsigned char*)(ws + OFF_LOW);
  unsigned char* mA = (unsigned char*)(ws + OFF_MA);
  unsigned char* mB = (unsigned char*)(ws + OFF_MB);
  float* red = (float*)(ws + OFF_RED);
  unsigned int* counts = (unsigned int*)(ws + OFF_CNT);

  // ---- surfaceness over sigma stack ----
  for (int si = 0; si < 2; ++si) {
    const float sg = (si == 0) ? 1.0f : 2.0f;
    if (si == 0) {
      conv_x_wmma<3><<<CONVBLOCKS, 128, 0, stream>>>(vin, tmpA, sg);
      conv_r_wmma<3><<<CONVBLOCKS, 128, 0, stream>>>(tmpA, tmpB, sg, NN, NN2);  // y
      conv_r_wmma<3><<<CONVBLOCKS, 128, 0, stream>>>(tmpB, tmpA, sg, NN2, NN);  // z
    } else {
      conv_x_wmma<6><<<CONVBLOCKS, 128, 0, stream>>>(vin, tmpA, sg);
      conv_r_wmma<6><<<CONVBLOCKS, 128, 0, stream>>>(tmpA, tmpB, sg, NN, NN2);
      conv_r_wmma<6><<<CONVBLOCKS, 128, 0, stream>>>(tmpB, tmpA, sg, NN2, NN);
    }
    surf_kernel<<<VOXBLOCKS, 256, 0, stream>>>(tmpA, tmpB, sg * sg);
    zero_u32_kernel<<<1, 64, 0, stream>>>((unsigned int*)red, 2);
    reduce_max_kernel<<<VOXBLOCKS, 256, 0, stream>>>(tmpB, red);
    combine_kernel<<<VOXBLOCKS, 256, 0, stream>>>(tmpB, red, smax, si == 0);
  }

  // ---- thresholds ----
  threshold_kernel<<<VOXBLOCKS, 256, 0, stream>>>(vin, smax, low, mA);

  // ---- hysteresis (256 Jacobi steps, ping-pong; even count ends in mA) ----
  for (int i = 0; i < 256; ++i) {
    const unsigned* s = (const unsigned*)((i & 1) ? mB : mA);
    unsigned* d = (unsigned*)((i & 1) ? mA : mB);
    hyst4_kernel<<<WORDBLOCKS, 256, 0, stream>>>((const unsigned*)low, s, d);
  }

  // ---- closing along z (xy radius is 0) ----
  morph_z4_kernel<<<WORDBLOCKS, 256, 0, stream>>>((const unsigned*)mA, (unsigned*)mB, 1);
  morph_z4_kernel<<<WORDBLOCKS, 256, 0, stream>>>((const unsigned*)mB, (unsigned*)mA, 0);

  // ---- connected components (labels reuse tmpA/tmpB) ----
  int* labA = (int*)tmpA;
  int* labB = (int*)tmpB;
  cc_init_kernel<<<VOXBLOCKS, 256, 0, stream>>>(mA, labA);
  for (int i = 0; i < 256; ++i) {
    const int* s = (i & 1) ? labB : labA;
    int* d = (i & 1) ? labA : labB;
    cc_step_kernel<<<VOXBLOCKS, 256, 0, stream>>>(mA, s, d);
  }

  // ---- size filter ----
  const int ncounts = NVOL * (VOX + 1);
  zero_u32_kernel<<<(ncounts + 255) / 256, 256, 0, stream>>>(counts, ncounts);
  cc_count_kernel<<<VOXBLOCKS, 256, 0, stream>>>(mA, labA, counts);
  final_kernel<<<VOXBLOCKS, 256, 0, stream>>>(mA, labA, counts, out);
}